// AdaEquiLayerNorm_86749749444860
// MI455X (gfx1250) — compile-verified
//
#include <hip/hip_runtime.h>
#include <hip/hip_bf16.h>
#include <math.h>

typedef __attribute__((ext_vector_type(2))) float v2f;
typedef __attribute__((ext_vector_type(8))) float v8f;

#define FREQ_DIM     256
#define TIME_DIM     512
#define MOD_DIM      352   // NUM_FEATURES + NUM_SCALAR
#define NUM_FEATURES 224
#define ROW_DIM      480
#define EPS          1e-5f

__device__ __forceinline__ float silu_f(float x) {
    return x / (1.0f + __expf(-x));
}

// ---------------------------------------------------------------------------
// Stage A.0: timestep embedding  emb[b, j] = cos/sin(t[b] * freq[j%128])
// ---------------------------------------------------------------------------
__global__ void emb_kernel(const float* __restrict__ t, float* __restrict__ emb, int B) {
    int idx = blockIdx.x * blockDim.x + threadIdx.x;
    if (idx >= B * FREQ_DIM) return;
    int b = idx / FREQ_DIM;
    int j = idx - b * FREQ_DIM;
    const int half = FREQ_DIM / 2;
    int k = (j < half) ? j : (j - half);
    // exp(-ln(10000) * k / 128)
    float freq = __expf(-9.210340371976184f * (float)k * (1.0f / (float)half));
    float arg = t[b] * freq;
    emb[idx] = (j < half) ? cosf(arg) : sinf(arg);
}

// ---------------------------------------------------------------------------
// Stage A.1-3: f32 WMMA GEMM, one wave per 16x16 tile.
//   D = act_out( act_in(A) @ B + bias )
// A: MxK row-major, B: KxN row-major, bias: N.
// M, N multiples of 16; K multiple of 4. EXEC stays all-1s per wave.
// ---------------------------------------------------------------------------
template <bool SILU_IN, bool SILU_OUT>
__global__ void wmma_gemm_kernel(const float* __restrict__ A,
                                 const float* __restrict__ Bm,
                                 const float* __restrict__ bias,
                                 float* __restrict__ D,
                                 int M, int N, int K) {
    const int lane = threadIdx.x & 31;
    const int wave = threadIdx.x >> 5;
    const int ntiles = N >> 4;
    const int totalTiles = (M >> 4) * ntiles;
    const int tile = blockIdx.x * (blockDim.x >> 5) + wave;
    if (tile >= totalTiles) return;              // wave-uniform exit

    const int mt = tile / ntiles;
    const int nt = tile - mt * ntiles;

    // A 16x4 layout: lane holds row (lane&15), K-pair base (lane>>4)*2
    // B 4x16 layout: lane holds col (lane&15), same K-pair split
    const int rc = lane & 15;
    const int kh = (lane >> 4) << 1;

    const float* __restrict__ arow = A + (size_t)(mt * 16 + rc) * K;
    const float* __restrict__ bcol = Bm + (nt * 16 + rc);

    v8f acc = {};
    for (int k = 0; k < K; k += 4) {
        float a0 = arow[k + kh + 0];
        float a1 = arow[k + kh + 1];
        if (SILU_IN) { a0 = silu_f(a0); a1 = silu_f(a1); }
        v2f a; a.x = a0; a.y = a1;
        v2f b;
        b.x = bcol[(size_t)(k + kh + 0) * N];
        b.y = bcol[(size_t)(k + kh + 1) * N];
        acc = __builtin_amdgcn_wmma_f32_16x16x4_f32(
            /*neg_a=*/false, a, /*neg_b=*/false, b,
            /*c_mod=*/(short)0, acc, /*reuse_a=*/false, /*reuse_b=*/false);
    }

    // C/D layout: VGPR r -> M = r (lanes 0-15) or r+8 (lanes 16-31); N = lane&15
    const int colg = nt * 16 + rc;
    const float bv = bias[colg];
    #pragma unroll
    for (int r = 0; r < 8; ++r) {
        int mrow = (lane < 16) ? r : (r + 8);
        float v = acc[r] + bv;
        if (SILU_OUT) v = silu_f(v);
        D[(size_t)(mt * 16 + mrow) * N + colg] = v;
    }
}

// ---------------------------------------------------------------------------
// Stage B: per-node adaptive equivariant layernorm. One wave32 per node.
// Row = 120 float4; segments: f4[0,32)=l0(128f), f4[32,80)=l1(192f),
// f4[80,120)=l2(160f). Single read + single write of the 384 MB stream.
// ---------------------------------------------------------------------------
__global__ void node_norm_kernel(const float* __restrict__ x,
                                 const int* __restrict__ batch,
                                 const float* __restrict__ modB,
                                 float* __restrict__ out, int N) {
    const int lane = threadIdx.x & 31;
    const int wave = threadIdx.x >> 5;
    const int node = blockIdx.x * (blockDim.x >> 5) + wave;
    if (node >= N) return;

    const float4* __restrict__ row  = (const float4*)(x   + (size_t)node * ROW_DIM);
    float4* __restrict__       orow = (float4*)      (out + (size_t)node * ROW_DIM);

    float4 v0 = row[lane];        // seg0 (scalar l=0)
    float4 v1 = row[32 + lane];   // seg1 (l=1)
    float4 v2 = row[64 + lane];   // seg1 if lane<16 else seg2
    float4 v3 = make_float4(0.f, 0.f, 0.f, 0.f);
    const bool has3 = (lane < 24);
    if (has3) v3 = row[96 + lane];  // seg2

    float s0  = v0.x + v0.y + v0.z + v0.w;
    float ss0 = v0.x*v0.x + v0.y*v0.y + v0.z*v0.z + v0.w*v0.w;
    float ss1 = v1.x*v1.x + v1.y*v1.y + v1.z*v1.z + v1.w*v1.w;
    float q2  = v2.x*v2.x + v2.y*v2.y + v2.z*v2.z + v2.w*v2.w;
    float ss2 = v3.x*v3.x + v3.y*v3.y + v3.z*v3.z + v3.w*v3.w;
    if (lane < 16) ss1 += q2; else ss2 += q2;

    #pragma unroll
    for (int off = 16; off > 0; off >>= 1) {
        s0  += __shfl_xor(s0,  off, 32);
        ss0 += __shfl_xor(ss0, off, 32);
        ss1 += __shfl_xor(ss1, off, 32);
        ss2 += __shfl_xor(ss2, off, 32);
    }

    const float mean = s0 * (1.0f / 128.0f);
    const float var  = ss0 * (1.0f / 128.0f) - mean * mean;
    const float r0 = rsqrtf(var + EPS);
    const float r1 = rsqrtf(ss1 * (1.0f / 192.0f) + EPS);
    const float r2 = rsqrtf(ss2 * (1.0f / 160.0f) + EPS);

    const int b = batch[node];
    const float* __restrict__ mrow = modB + (size_t)b * MOD_DIM;
    const float sc0 = (1.0f + mrow[0]) * r0;
    const float sc1 = (1.0f + mrow[1]) * r1;
    const float sc2 = (1.0f + mrow[2]) * r2;
    const float4 sh = *(const float4*)(mrow + NUM_FEATURES + 4 * lane);

    float4 o0;
    o0.x = (v0.x - mean) * sc0 + sh.x;
    o0.y = (v0.y - mean) * sc0 + sh.y;
    o0.z = (v0.z - mean) * sc0 + sh.z;
    o0.w = (v0.w - mean) * sc0 + sh.w;
    orow[lane] = o0;

    orow[32 + lane] = make_float4(v1.x * sc1, v1.y * sc1, v1.z * sc1, v1.w * sc1);

    const float sm = (lane < 16) ? sc1 : sc2;
    orow[64 + lane] = make_float4(v2.x * sm, v2.y * sm, v2.z * sm, v2.w * sm);

    if (has3)
        orow[96 + lane] = make_float4(v3.x * sc2, v3.y * sc2, v3.z * sc2, v3.w * sc2);
}

// ---------------------------------------------------------------------------
extern "C" void kernel_launch(void* const* d_in, const int* in_sizes, int n_in,
                              void* d_out, int out_size, void* d_ws, size_t ws_size,
                              hipStream_t stream) {
    const float* node_input = (const float*)d_in[0];
    const float* t          = (const float*)d_in[1];
    const int*   batch      = (const int*)  d_in[2];
    const float* w1         = (const float*)d_in[3];
    const float* b1         = (const float*)d_in[4];
    const float* w2         = (const float*)d_in[5];
    const float* b2         = (const float*)d_in[6];
    const float* wm         = (const float*)d_in[7];
    const float* bm         = (const float*)d_in[8];
    float* out = (float*)d_out;

    const int N = in_sizes[0] / ROW_DIM;   // 100000 nodes
    const int B = in_sizes[1];             // 1024 timesteps

    // workspace layout (floats)
    float* ws   = (float*)d_ws;
    float* emb  = ws;                                   // B * 256
    float* h1   = emb  + (size_t)B * FREQ_DIM;          // B * 512
    float* temb = h1   + (size_t)B * TIME_DIM;          // B * 512
    float* modB = temb + (size_t)B * TIME_DIM;          // B * 352

    // A.0: timestep embedding
    {
        int total = B * FREQ_DIM;
        emb_kernel<<<(total + 255) / 256, 256, 0, stream>>>(t, emb, B);
    }
    // A.1: h1 = silu(emb @ w1 + b1)            (B x 512, K=256)
    {
        int tiles = (B / 16) * (TIME_DIM / 16);
        wmma_gemm_kernel<false, true><<<(tiles + 3) / 4, 128, 0, stream>>>(
            emb, w1, b1, h1, B, TIME_DIM, FREQ_DIM);
    }
    // A.2: temb = h1 @ w2 + b2                 (B x 512, K=512)
    {
        int tiles = (B / 16) * (TIME_DIM / 16);
        wmma_gemm_kernel<false, false><<<(tiles + 3) / 4, 128, 0, stream>>>(
            h1, w2, b2, temb, B, TIME_DIM, TIME_DIM);
    }
    // A.3: modB = silu(temb) @ wm + bm         (B x 352, K=512)
    {
        int tiles = (B / 16) * (MOD_DIM / 16);
        wmma_gemm_kernel<true, false><<<(tiles + 3) / 4, 128, 0, stream>>>(
            temb, wm, bm, modB, B, MOD_DIM, TIME_DIM);
    }
    // B: streaming per-node norm (one wave per node, 8 waves per block)
    {
        const int wavesPerBlock = 8;
        int blocks = (N + wavesPerBlock - 1) / wavesPerBlock;
        node_norm_kernel<<<blocks, 32 * wavesPerBlock, 0, stream>>>(
            node_input, batch, modB, out, N);
    }
}